// BatchNeuralMemoryV3_19516331393469
// MI455X (gfx1250) — compile-verified
//
#include <hip/hip_runtime.h>
#include <math.h>

// Problem constants (from reference setup_inputs)
#define BB 4
#define SS 512
#define DD 512
#define II 1024
#define CC 32
#define NCHUNK 16
#define EPSR 1e-6f
#define BASE_LRF 0.01f

typedef float v2f __attribute__((ext_vector_type(2)));
typedef float v4f __attribute__((ext_vector_type(4)));
typedef float v8f __attribute__((ext_vector_type(8)));
typedef int i32x4 __attribute__((vector_size(16)));   // matches builtin's V4i param

#define AS1 __attribute__((address_space(1)))
#define AS3 __attribute__((address_space(3)))

// ---- CDNA5 async global->LDS copies (ASYNCcnt path), with portable fallback
#if defined(__has_builtin)
#if __has_builtin(__builtin_amdgcn_global_load_async_to_lds_b128) && \
    __has_builtin(__builtin_amdgcn_global_load_async_to_lds_b32) &&  \
    __has_builtin(__builtin_amdgcn_s_wait_asynccnt)
#define USE_ASYNC 1
#endif
#endif
#ifndef USE_ASYNC
#define USE_ASYNC 0
#endif

__device__ __forceinline__ void async_cp16(float* l, const float* g) {
#if USE_ASYNC
    __builtin_amdgcn_global_load_async_to_lds_b128(
        (AS1 i32x4*)(void*)g, (AS3 i32x4*)(void*)l, 0, 0);
#else
    *(v4f*)l = *(const v4f*)g;
#endif
}
__device__ __forceinline__ void async_cp4(float* l, const float* g) {
#if USE_ASYNC
    __builtin_amdgcn_global_load_async_to_lds_b32(
        (AS1 int*)(void*)g, (AS3 int*)(void*)l, 0, 0);
#else
    *l = *g;
#endif
}
__device__ __forceinline__ void async_join() {
#if USE_ASYNC
    __builtin_amdgcn_s_wait_asynccnt(0);
#endif
}

__device__ __forceinline__ float sigf(float x) { return 1.0f / (1.0f + __expf(-x)); }
__device__ __forceinline__ float siluf(float x) { return x * sigf(x); }
__device__ __forceinline__ float dsiluf(float x) {
    float s = sigf(x);
    return s * (1.0f + x * (1.0f - s));
}

__device__ __forceinline__ float block_reduce(float v, float* red, int tid) {
    red[tid] = v;
    __syncthreads();
    for (int s = 128; s > 0; s >>= 1) {
        if (tid < s) red[tid] += red[tid + s];
        __syncthreads();
    }
    float r = red[0];
    __syncthreads();
    return r;
}

// ---------------------------------------------------------------------------
// Generic tiled WMMA GEMM: C[M,N] = epi( sum_k A[m,k] * B(k,n) )
//   A row-major [M,K] (lda). B: transB ? stored [N,K] (ldb) : stored [K,N] (ldb)
//   modes: 0 = raw store, 1 = silu, 2 = stash raw in C2 + silu in C,
//          3 = C = acc * dsilu(C2)
// LDS tiles use k-contiguous rows (row = KT+4 floats = 144B, keeps 16B align
// for b128 async stores and gives ds_load_b64 fragment reads).
// ---------------------------------------------------------------------------
#define TM 32
#define TN 64
#define KT 32
#define KP 4

__global__ __launch_bounds__(256)
void gemm_wmma(const float* __restrict__ A, const float* __restrict__ B,
               float* __restrict__ C, float* __restrict__ C2,
               int M, int N, int K, int lda, int ldb, int ldc,
               long sA, long sB, long sC,
               int transB, int mode)
{
    __shared__ __align__(16) float As[TM][KT + KP];
    __shared__ __align__(16) float Bs[TN][KT + KP];
    const int tid = threadIdx.x;
    const int bz = blockIdx.z;
    A += (long)bz * sA; B += (long)bz * sB; C += (long)bz * sC;
    if (C2) C2 += (long)bz * sC;

    const int m0 = blockIdx.y * TM;
    const int n0 = blockIdx.x * TN;
    const int lane = tid & 31, wid = tid >> 5;
    const int wr = wid >> 2, wc = wid & 3;       // wave tile row (0..1) / col (0..3)
    const int mf = lane & 15;                    // M (or N) index within fragment
    const int kh = (lane >> 4) << 1;             // K half offset: 0 or 2

    v8f acc = {0.f, 0.f, 0.f, 0.f, 0.f, 0.f, 0.f, 0.f};

    for (int k0 = 0; k0 < K; k0 += KT) {
        if (k0 + KT < K)  // speculative next-tile prefetch -> global_prefetch_b8
            __builtin_prefetch(&A[(long)(m0 + (tid >> 5)) * lda + k0 + KT], 0, 1);
        // A tile: 32x32, one 16B async transfer per thread (k-contiguous both sides)
        {
            int m = tid >> 3, k4 = (tid & 7) << 2;
            async_cp16(&As[m][k4], &A[(long)(m0 + m) * lda + (k0 + k4)]);
        }
        if (transB) {  // B stored [N,K]: k-contiguous -> 16B async transfers
            #pragma unroll
            for (int r = 0; r < 2; ++r) {
                int idx = tid + r * 256;
                int n = idx >> 3, k4 = (idx & 7) << 2;
                async_cp16(&Bs[n][k4], &B[(long)(n0 + n) * ldb + (k0 + k4)]);
            }
        } else {       // B stored [K,N]: transpose staging, elementwise async b32
            #pragma unroll
            for (int r = 0; r < 8; ++r) {
                int idx = tid + r * 256;
                int n = idx & 63, k = idx >> 6;
                async_cp4(&Bs[n][k], &B[(long)(k0 + k) * ldb + (n0 + n)]);
            }
        }
        async_join();
        __syncthreads();
        #pragma unroll
        for (int kk = 0; kk < KT; kk += 4) {
            v2f a = *(const v2f*)&As[wr * 16 + mf][kk + kh];
            v2f b = *(const v2f*)&Bs[wc * 16 + mf][kk + kh];
            acc = __builtin_amdgcn_wmma_f32_16x16x4_f32(false, a, false, b,
                                                        (short)0, acc, false, false);
        }
        __syncthreads();
    }

    const int row0 = m0 + wr * 16 + (lane >> 4) * 8;  // C/D layout: VGPR r -> rows r, r+8
    const int col = n0 + wc * 16 + mf;
    #pragma unroll
    for (int r = 0; r < 8; ++r) {
        long off = (long)(row0 + r) * ldc + col;
        float v = acc[r];
        if (mode == 1)      C[off] = siluf(v);
        else if (mode == 2) { C2[off] = v; C[off] = siluf(v); }
        else if (mode == 3) C[off] = v * dsiluf(C2[off]);
        else                C[off] = v;
    }
}

// ---------------------------------------------------------------------------
// Fused rank-32 state update (dual accumulator, scan scaling folded into B):
//   S_new = aS*S_old + GA^T diag(cS) GB ;  W_new = aM*W_old + f0*S_old + GA^T diag(cM) GB
// GA: [CC, Mdim] (lda), GB: [CC, Ndim] (ldb). One pass over old state.
// ---------------------------------------------------------------------------
__global__ __launch_bounds__(256)
void update_wmma(const float* __restrict__ GA, const float* __restrict__ GB,
                 float* __restrict__ W, float* __restrict__ Smat,
                 const float* __restrict__ cS, const float* __restrict__ cM,
                 const float* __restrict__ scal,
                 int Mdim, int Ndim, int lda, int ldb,
                 long sGA, long sGB, long sW)
{
    __shared__ __align__(16) float As[TM][CC + KP];
    __shared__ __align__(16) float BsS[TN][CC + KP];
    __shared__ __align__(16) float BsM[TN][CC + KP];
    const int tid = threadIdx.x;
    const int b = blockIdx.z;
    GA += (long)b * sGA; GB += (long)b * sGB;
    W += (long)b * sW;  Smat += (long)b * sW;
    const float* cs = cS + b * CC;
    const float* cm = cM + b * CC;
    const float aS = scal[b * 4 + 0], aM = scal[b * 4 + 1], f0 = scal[b * 4 + 2];

    const int m0 = blockIdx.y * TM;
    const int n0 = blockIdx.x * TN;
    const int lane = tid & 31, wid = tid >> 5;
    const int wr = wid >> 2, wc = wid & 3;
    const int mf = lane & 15;
    const int kh = (lane >> 4) << 1;

    // Raw A^T tile (CC x TM), transpose staging via async b32 (m fast = coalesced)
    #pragma unroll
    for (int r = 0; r < (TM * CC) / 256; ++r) {
        int idx = tid + r * 256;
        int m = idx & 31, t = idx >> 5;
        async_cp4(&As[m][t], &GA[(long)t * lda + (m0 + m)]);
    }
    // Scaled B tiles (must route through VGPRs for the diag(c) scaling)
    #pragma unroll
    for (int r = 0; r < (CC * TN) / 256; ++r) {
        int idx = tid + r * 256;
        int n = idx & 63, t = idx >> 6;
        float v = GB[(long)t * ldb + (n0 + n)];
        BsS[n][t] = cs[t] * v;
        BsM[n][t] = cm[t] * v;
    }
    async_join();
    __syncthreads();

    v8f accS = {0.f, 0.f, 0.f, 0.f, 0.f, 0.f, 0.f, 0.f};
    v8f accM = {0.f, 0.f, 0.f, 0.f, 0.f, 0.f, 0.f, 0.f};
    #pragma unroll
    for (int kk = 0; kk < CC; kk += 4) {
        v2f a  = *(const v2f*)&As [wr * 16 + mf][kk + kh];
        v2f bs = *(const v2f*)&BsS[wc * 16 + mf][kk + kh];
        accS = __builtin_amdgcn_wmma_f32_16x16x4_f32(false, a, false, bs,
                                                     (short)0, accS, false, false);
        v2f bm = *(const v2f*)&BsM[wc * 16 + mf][kk + kh];
        accM = __builtin_amdgcn_wmma_f32_16x16x4_f32(false, a, false, bm,
                                                     (short)0, accM, false, false);
    }

    const int row0 = m0 + wr * 16 + (lane >> 4) * 8;
    const int col = n0 + wc * 16 + mf;
    #pragma unroll
    for (int r = 0; r < 8; ++r) {
        long off = (long)(row0 + r) * Ndim + col;
        float wold = W[off];
        float sold = Smat[off];
        W[off]    = aM * wold + f0 * sold + accM[r];
        Smat[off] = aS * sold + accS[r];
    }
}

// ---------------------------------------------------------------------------
// Row-wise RMS normalize in place: p[:] *= rsqrt(mean(p^2)+eps) * w[:]
// ---------------------------------------------------------------------------
__global__ __launch_bounds__(256)
void rms_rows(float* __restrict__ buf, const float* __restrict__ w)
{
    __shared__ float red[256];
    long row = blockIdx.x;
    float* p = buf + row * DD;
    int tid = threadIdx.x;
    float v0 = p[tid], v1 = p[tid + 256];
    float ss = block_reduce(v0 * v0 + v1 * v1, red, tid);
    float r = rsqrtf(ss / (float)DD + EPSR);
    p[tid]       = v0 * r * w[tid];
    p[tid + 256] = v1 * r * w[tid + 256];
}

// ---------------------------------------------------------------------------
// Gates: theta = lr*sig(x.tw), alpha = sig(x.aw), eta = sig(x.ew) per token row
// ---------------------------------------------------------------------------
__global__ __launch_bounds__(128)
void gates_kernel(const float* __restrict__ x, const float* __restrict__ aw,
                  const float* __restrict__ tw, const float* __restrict__ ew,
                  float* __restrict__ alphaA, float* __restrict__ thetaA,
                  float* __restrict__ etaA)
{
    __shared__ float r0[128], r1[128], r2[128];
    long row = blockIdx.x;
    const float* xr = x + row * DD;
    int tid = threadIdx.x;
    float sa = 0.f, st = 0.f, se = 0.f;
    for (int d = tid; d < DD; d += 128) {
        float xv = xr[d];
        sa += xv * aw[d]; st += xv * tw[d]; se += xv * ew[d];
    }
    r0[tid] = sa; r1[tid] = st; r2[tid] = se;
    __syncthreads();
    for (int s = 64; s > 0; s >>= 1) {
        if (tid < s) { r0[tid] += r0[tid + s]; r1[tid] += r1[tid + s]; r2[tid] += r2[tid + s]; }
        __syncthreads();
    }
    if (tid == 0) {
        alphaA[row] = sigf(r0[0]);
        thetaA[row] = BASE_LRF * sigf(r1[0]);
        etaA[row]   = sigf(r2[0]);
    }
}

// ---------------------------------------------------------------------------
// Per-token RMSNorm backward: from O (raw W2@h) produce d_o, gln, and norms.
// ---------------------------------------------------------------------------
__global__ __launch_bounds__(256)
void chunk_bwd(const float* __restrict__ Ob, const float* __restrict__ Kb,
               const float* __restrict__ Vb, const float* __restrict__ LN,
               const float* __restrict__ thetaA,
               float* __restrict__ DO_, float* __restrict__ GLN,
               float* __restrict__ ssln, float* __restrict__ ndo, int chunk)
{
    __shared__ float red[256];
    int t = blockIdx.x, b = blockIdx.y, tid = threadIdx.x;
    long rc = (long)b * CC + t;
    long rs = (long)b * SS + (long)chunk * CC + t;
    const float* o = Ob + rc * DD;
    const float* k = Kb + rs * DD;
    const float* v = Vb + rs * DD;
    const float* ln = LN + (long)b * DD;
    float th = thetaA[rs];
    int d0 = tid, d1 = tid + 256;
    float o0 = o[d0], o1 = o[d1];
    float ss = block_reduce(o0 * o0 + o1 * o1, red, tid);
    float r = rsqrtf(ss / (float)DD + EPSR);
    float oh0 = o0 * r, oh1 = o1 * r;
    float l0 = ln[d0], l1 = ln[d1];
    float sc = 2.0f * th / (float)DD;                 // dL/dpred = 2*(pred-v)*theta/D
    float dp0 = sc * ((k[d0] + oh0 * l0) - v[d0]);
    float dp1 = sc * ((k[d1] + oh1 * l1) - v[d1]);
    float g0 = dp0 * oh0, g1 = dp1 * oh1;             // grad wrt ln
    GLN[rc * DD + d0] = g0; GLN[rc * DD + d1] = g1;
    float u0 = dp0 * l0, u1 = dp1 * l1;
    float dot = block_reduce(u0 * o0 + u1 * o1, red, tid);
    float c3 = r * r * r * dot / (float)DD;           // rmsnorm backward
    float do0 = r * u0 - c3 * o0, do1 = r * u1 - c3 * o1;
    DO_[rc * DD + d0] = do0; DO_[rc * DD + d1] = do1;
    float sg = block_reduce(g0 * g0 + g1 * g1, red, tid);
    float sd = block_reduce(do0 * do0 + do1 * do1, red, tid);
    if (tid == 0) { ssln[rc] = sg; ndo[rc] = sd; }
}

// ---------------------------------------------------------------------------
// Global grad-norm clip per (b,t): ||g||^2 = |dpre|^2|k|^2 + |d_o|^2|h|^2 + |gln|^2
// ---------------------------------------------------------------------------
__global__ __launch_bounds__(256)
void clip_kernel(const float* __restrict__ DPRE, const float* __restrict__ Hb,
                 const float* __restrict__ Kb, const float* __restrict__ ssln,
                 const float* __restrict__ ndo, float* __restrict__ clipA, int chunk)
{
    __shared__ float red[256];
    int t = blockIdx.x, b = blockIdx.y, tid = threadIdx.x;
    long rc = (long)b * CC + t;
    long rs = (long)b * SS + (long)chunk * CC + t;
    const float* dp = DPRE + rc * II;
    const float* h = Hb + rc * II;
    const float* k = Kb + rs * DD;
    float sdp = 0.f, sh = 0.f, sk = 0.f;
    for (int i = tid; i < II; i += 256) { float a = dp[i]; sdp += a * a; float c = h[i]; sh += c * c; }
    for (int i = tid; i < DD; i += 256) { float a = k[i]; sk += a * a; }
    sdp = block_reduce(sdp, red, tid);
    sh  = block_reduce(sh, red, tid);
    sk  = block_reduce(sk, red, tid);
    if (tid == 0) {
        float tot = sdp * sk + ndo[rc] * sh + ssln[rc];
        clipA[rc] = fminf(1.0f / (sqrtf(tot) + 1e-6f), 1.0f);  // MAX_GRAD_NORM = 1
    }
}

// ---------------------------------------------------------------------------
// Scalar scan coefficients (closed form of the two associative scans).
// ---------------------------------------------------------------------------
__global__ void scan_coef(const float* __restrict__ etaA, const float* __restrict__ alphaA,
                          const float* __restrict__ clipA,
                          float* __restrict__ cS, float* __restrict__ cM,
                          float* __restrict__ scal, int chunk)
{
    int b = blockIdx.x;
    if (threadIdx.x != 0) return;
    float eta[CC], beta[CC], cl[CC], wS[CC], Bsuf[CC], Ein[CC];
    long base = (long)b * SS + (long)chunk * CC;
    for (int t = 0; t < CC; ++t) {
        eta[t] = etaA[base + t];
        beta[t] = 1.0f - alphaA[base + t];
        cl[t] = clipA[b * CC + t];
    }
    float p = 1.f;
    for (int t = CC - 1; t >= 0; --t) { wS[t] = p; p *= eta[t]; }
    float aS = p;
    p = 1.f;
    for (int t = CC - 1; t >= 0; --t) { Bsuf[t] = p; p *= beta[t]; }
    float aM = p;
    p = 1.f;
    for (int t = 0; t < CC; ++t) { p *= eta[t]; Ein[t] = p; }
    float f0 = 0.f;
    for (int t = 0; t < CC; ++t) f0 += Bsuf[t] * Ein[t];
    for (int t = 0; t < CC; ++t) {
        float wM = 0.f, e = 1.f;
        for (int s = t; s < CC; ++s) { if (s > t) e *= eta[s]; wM += Bsuf[s] * e; }
        cS[b * CC + t] = -cl[t] * wS[t];
        cM[b * CC + t] = -cl[t] * wM;
    }
    scal[b * 4 + 0] = aS; scal[b * 4 + 1] = aM; scal[b * 4 + 2] = f0;
}

// ---------------------------------------------------------------------------
// ln / S_ln state update (vector-sized).
// ---------------------------------------------------------------------------
__global__ __launch_bounds__(256)
void ln_update(float* __restrict__ LN, float* __restrict__ SLN,
               const float* __restrict__ GLN, const float* __restrict__ cS,
               const float* __restrict__ cM, const float* __restrict__ scal)
{
    int b = blockIdx.x, tid = threadIdx.x;
    float aS = scal[b * 4 + 0], aM = scal[b * 4 + 1], f0 = scal[b * 4 + 2];
    for (int d = tid; d < DD; d += 256) {
        float gS = 0.f, gM = 0.f;
        for (int t = 0; t < CC; ++t) {
            float g = GLN[((long)b * CC + t) * DD + d];
            gS += cS[b * CC + t] * g;
            gM += cM[b * CC + t] * g;
        }
        float so = SLN[(long)b * DD + d], wo = LN[(long)b * DD + d];
        LN[(long)b * DD + d]  = aM * wo + f0 * so + gM;
        SLN[(long)b * DD + d] = aS * so + gS;
    }
}

// ---------------------------------------------------------------------------
// Final: out = q + rms(o2) * ln[b]
// ---------------------------------------------------------------------------
__global__ __launch_bounds__(256)
void final_out(const float* __restrict__ Qb, const float* __restrict__ O2,
               const float* __restrict__ LN, float* __restrict__ out)
{
    __shared__ float red[256];
    long row = blockIdx.x;
    int b = (int)(row / SS);
    int tid = threadIdx.x;
    const float* o = O2 + row * DD;
    const float* q = Qb + row * DD;
    const float* ln = LN + (long)b * DD;
    float o0 = o[tid], o1 = o[tid + 256];
    float ss = block_reduce(o0 * o0 + o1 * o1, red, tid);
    float r = rsqrtf(ss / (float)DD + EPSR);
    out[row * DD + tid]       = q[tid] + o0 * r * ln[tid];
    out[row * DD + tid + 256] = q[tid + 256] + o1 * r * ln[tid + 256];
}

// Broadcast initial params per batch, zero the surprise state.
__global__ void bcast_init(float* __restrict__ dst, float* __restrict__ sdst,
                           const float* __restrict__ src, long nper, long ntot)
{
    long i = (long)blockIdx.x * blockDim.x + threadIdx.x;
    if (i < ntot) { dst[i] = src[i % nper]; sdst[i] = 0.f; }
}

// ---------------------------------------------------------------------------
extern "C" void kernel_launch(void* const* d_in, const int* in_sizes, int n_in,
                              void* d_out, int out_size, void* d_ws, size_t ws_size,
                              hipStream_t stream)
{
    const float* x   = (const float*)d_in[0];
    const float* wq  = (const float*)d_in[1];
    const float* wk  = (const float*)d_in[2];
    const float* wv  = (const float*)d_in[3];
    const float* qn  = (const float*)d_in[4];
    const float* kn  = (const float*)d_in[5];
    const float* aw  = (const float*)d_in[6];
    const float* tw  = (const float*)d_in[7];
    const float* ew  = (const float*)d_in[8];
    const float* mw1 = (const float*)d_in[9];
    const float* mw2 = (const float*)d_in[10];
    const float* mln = (const float*)d_in[11];
    float* out = (float*)d_out;

    float* p = (float*)d_ws;
    auto alloc = [&](long n) { float* r = p; p += n; return r; };
    float* W1   = alloc((long)BB * II * DD);
    float* W2   = alloc((long)BB * DD * II);
    float* LN   = alloc((long)BB * DD);
    float* S1   = alloc((long)BB * II * DD);
    float* S2   = alloc((long)BB * DD * II);
    float* SLN  = alloc((long)BB * DD);
    float* Kb   = alloc((long)BB * SS * DD);
    float* Vb   = alloc((long)BB * SS * DD);
    float* Qb   = alloc((long)BB * SS * DD);
    float* thetaA = alloc(BB * SS);
    float* alphaA = alloc(BB * SS);
    float* etaA   = alloc(BB * SS);
    float* PRE  = alloc((long)BB * CC * II);
    float* Hb   = alloc((long)BB * CC * II);
    float* Ob   = alloc((long)BB * CC * DD);
    float* DO_  = alloc((long)BB * CC * DD);
    float* GLN  = alloc((long)BB * CC * DD);
    float* DPRE = alloc((long)BB * CC * II);
    float* ssln = alloc(BB * CC);
    float* ndo  = alloc(BB * CC);
    float* clipA = alloc(BB * CC);
    float* cS   = alloc(BB * CC);
    float* cM   = alloc(BB * CC);
    float* scal = alloc(BB * 4);
    float* H2   = alloc((long)BB * SS * II);
    float* O2   = alloc((long)BB * SS * DD);
    (void)in_sizes; (void)n_in; (void)out_size; (void)ws_size;

    // ---- init state: W=param broadcast, S=0
    { long n = (long)BB * II * DD; bcast_init<<<dim3((n + 255) / 256), 256, 0, stream>>>(W1, S1, mw1, (long)II * DD, n); }
    { long n = (long)BB * DD * II; bcast_init<<<dim3((n + 255) / 256), 256, 0, stream>>>(W2, S2, mw2, (long)DD * II, n); }
    { long n = (long)BB * DD;      bcast_init<<<dim3((n + 255) / 256), 256, 0, stream>>>(LN, SLN, mln, (long)DD, n); }

    const int MBS = BB * SS;
    // ---- projections: k = rms(silu(x wk^T)), v = silu(x wv^T), q = rms(silu(x wq^T))
    gemm_wmma<<<dim3(DD / TN, MBS / TM, 1), 256, 0, stream>>>(x, wk, Kb, nullptr,
        MBS, DD, DD, DD, DD, DD, 0, 0, 0, /*transB=*/1, /*mode=*/1);
    rms_rows<<<dim3(MBS), 256, 0, stream>>>(Kb, kn);
    gemm_wmma<<<dim3(DD / TN, MBS / TM, 1), 256, 0, stream>>>(x, wv, Vb, nullptr,
        MBS, DD, DD, DD, DD, DD, 0, 0, 0, 1, 1);
    gemm_wmma<<<dim3(DD / TN, MBS / TM, 1), 256, 0, stream>>>(x, wq, Qb, nullptr,
        MBS, DD, DD, DD, DD, DD, 0, 0, 0, 1, 1);
    rms_rows<<<dim3(MBS), 256, 0, stream>>>(Qb, qn);
    gates_kernel<<<dim3(MBS), 128, 0, stream>>>(x, aw, tw, ew, alphaA, thetaA, etaA);

    // ---- sequential chunk scan
    for (int c = 0; c < NCHUNK; ++c) {
        const float* Kc = Kb + (long)c * CC * DD;  // per-batch stride SS*DD below
        // PRE = Kc @ W1^T ; H = silu(PRE)
        gemm_wmma<<<dim3(II / TN, CC / TM, BB), 256, 0, stream>>>(Kc, W1, Hb, PRE,
            CC, II, DD, DD, DD, II, (long)SS * DD, (long)II * DD, (long)CC * II, 1, 2);
        // O = H @ W2^T
        gemm_wmma<<<dim3(DD / TN, CC / TM, BB), 256, 0, stream>>>(Hb, W2, Ob, nullptr,
            CC, DD, II, II, II, DD, (long)CC * II, (long)DD * II, (long)CC * DD, 1, 0);
        // rmsnorm backward per token -> d_o, gln, norms
        chunk_bwd<<<dim3(CC, BB), 256, 0, stream>>>(Ob, Kb, Vb, LN, thetaA, DO_, GLN, ssln, ndo, c);
        // dpre = (d_o @ W2) * dsilu(PRE)
        gemm_wmma<<<dim3(II / TN, CC / TM, BB), 256, 0, stream>>>(DO_, W2, DPRE, PRE,
            CC, II, DD, DD, II, II, (long)CC * DD, (long)DD * II, (long)CC * II, 0, 3);
        clip_kernel<<<dim3(CC, BB), 256, 0, stream>>>(DPRE, Hb, Kb, ssln, ndo, clipA, c);
        scan_coef<<<dim3(BB), 32, 0, stream>>>(etaA, alphaA, clipA, cS, cM, scal, c);
        // W1/S1 <- rank-32 update: GA=DPRE [CC,II], GB=Kc [CC,DD]
        update_wmma<<<dim3(DD / TN, II / TM, BB), 256, 0, stream>>>(DPRE, Kc, W1, S1,
            cS, cM, scal, II, DD, II, DD, (long)CC * II, (long)SS * DD, (long)II * DD);
        // W2/S2 <- rank-32 update: GA=DO_ [CC,DD], GB=H [CC,II]
        update_wmma<<<dim3(II / TN, DD / TM, BB), 256, 0, stream>>>(DO_, Hb, W2, S2,
            cS, cM, scal, DD, II, DD, II, (long)CC * DD, (long)CC * II, (long)DD * II);
        ln_update<<<dim3(BB), 256, 0, stream>>>(LN, SLN, GLN, cS, cM, scal);
    }

    // ---- retrieval from final memory
    gemm_wmma<<<dim3(II / TN, SS / TM, BB), 256, 0, stream>>>(Qb, W1, H2, nullptr,
        SS, II, DD, DD, DD, II, (long)SS * DD, (long)II * DD, (long)SS * II, 1, 1);
    gemm_wmma<<<dim3(DD / TN, SS / TM, BB), 256, 0, stream>>>(H2, W2, O2, nullptr,
        SS, DD, II, II, II, DD, (long)SS * II, (long)DD * II, (long)SS * DD, 1, 0);
    final_out<<<dim3(MBS), 256, 0, stream>>>(Qb, O2, LN, out);
}